// MambaDepthWiseEEG_23742579212692
// MI455X (gfx1250) — compile-verified
//
#include <hip/hip_runtime.h>
#include <math.h>

namespace {
constexpr int Bn = 16, Cn = 22, Ln = 1000, Hn = 128, DIN = 256, DS = 16, DC = 4, DTR = 8, NCn = 4;
constexpr int MIX  = Hn * Cn;   // 2816
constexpr int MIX3 = 3 * MIX;   // 8448
constexpr int KSEG1 = 4;        // 2816 = 4 * 11 * 64
constexpr int KSEG2 = 6;        // 8448 = 6 * 22 * 64
}

typedef __attribute__((ext_vector_type(2))) float v2f;
typedef __attribute__((ext_vector_type(8))) float v8f;

__device__ __forceinline__ v8f wmma_f32_16x16x4(v2f a, v2f b, v8f c) {
#if defined(__AMDGCN__) && __has_builtin(__builtin_amdgcn_wmma_f32_16x16x4_f32)
  return __builtin_amdgcn_wmma_f32_16x16x4_f32(false, a, false, b, (short)0, c, false, false);
#else
  (void)a; (void)b;
  return c;  // host pass only
#endif
}

#if defined(__AMDGCN__)
// Async global->LDS 16B/lane copy (CDNA5, tracked by ASYNCcnt). GVS addressing:
// lds_byte = LDS dest address VGPR, goff = unsigned 32-bit byte offset VGPR, base = SGPR64.
__device__ __forceinline__ void async_copy_b128(unsigned int lds_byte,
                                                unsigned long long gbase,
                                                unsigned int goff) {
  asm volatile("global_load_async_to_lds_b128 %0, %1, %2"
               :: "v"(lds_byte), "v"(goff), "s"(gbase) : "memory");
}
__device__ __forceinline__ void wait_async0() {
  asm volatile("s_wait_asynccnt 0x0" ::: "memory");
}
#else
__device__ __forceinline__ void async_copy_b128(unsigned int, unsigned long long, unsigned int) {}
__device__ __forceinline__ void wait_async0() {}
#endif

// Flat pointer to __shared__ -> LDS byte offset (LDS aperture keeps offset in low 32 bits).
__device__ __forceinline__ unsigned int lds_off(const void* p) {
  return (unsigned int)(unsigned long long)(uintptr_t)p;
}

// One workgroup per (b,c). 256 threads: thread = inner channel d.
// Fuses: rank-1 input projection, depthwise conv + silu, x_dbl GEMM (WMMA f32),
// dt projection + softplus, selective-scan recurrence, silu(z) gate, mean over L.
__global__ __launch_bounds__(256)
void k_scan(const float* __restrict__ x, const float* __restrict__ W_emb,
            const float* __restrict__ b_emb, const float* __restrict__ in_proj_w,
            const float* __restrict__ conv_w, const float* __restrict__ conv_b,
            const float* __restrict__ x_proj_w, const float* __restrict__ dt_proj_w,
            const float* __restrict__ dt_proj_b, const float* __restrict__ A_log,
            const float* __restrict__ Dvec, float* __restrict__ y_mean)
{
  __shared__ float xs[Ln + 4];            // x row, 3 leading zeros (causal pad)
  __shared__ float xp[40][DIN + 4];       // x_proj_w[c], padded stride vs LDS banks
  __shared__ float u_buf[16][DIN + 4];    // silu(conv) chunk, [l][d]
  __shared__ float xdbl[16][52];          // x_dbl chunk, [l][q] (q<40 valid, padded to 48)

  const int bc = blockIdx.x;
  const int b  = bc / Cn;
  const int c  = bc % Cn;
  const int d  = threadIdx.x;

  for (int i = threadIdx.x; i < Ln + 3; i += 256)
    xs[i] = (i < 3) ? 0.f : x[(b * Cn + c) * Ln + (i - 3)];

  const float* xpw = x_proj_w + c * 40 * DIN;
  for (int i = threadIdx.x; i < 40 * DIN; i += 256)
    xp[i / DIN][i % DIN] = xpw[i];

  // rank-1 collapse of emb @ in_proj_w (exact): w_eff/b_eff for x-path and z-path
  const float* we  = W_emb + c * Hn;
  const float* be  = b_emb + c * Hn;
  const float* ipu = in_proj_w + (c * 2 * DIN + d) * Hn;
  const float* ipz = in_proj_w + (c * 2 * DIN + DIN + d) * Hn;
  float weff_u = 0.f, beff_u = 0.f, weff_z = 0.f, beff_z = 0.f;
  for (int h = 0; h < Hn; ++h) {
    const float wh = we[h], bh = be[h];
    weff_u += wh * ipu[h]; beff_u += bh * ipu[h];
    weff_z += wh * ipz[h]; beff_z += bh * ipz[h];
  }
  float cw[DC];
  for (int k = 0; k < DC; ++k) cw[k] = conv_w[(c * DIN + d) * DC + k];
  const float cb = conv_b[c * DIN + d];
  float dtw[DTR];
  for (int r = 0; r < DTR; ++r) dtw[r] = dt_proj_w[(c * DIN + d) * DTR + r];
  const float dtb = dt_proj_b[c * DIN + d];
  float Amat[DS];
  for (int s = 0; s < DS; ++s) Amat[s] = -__expf(A_log[(c * DIN + d) * DS + s]);
  const float Dd = Dvec[c * DIN + d];

  float hst[DS];
  for (int s = 0; s < DS; ++s) hst[s] = 0.f;
  float acc = 0.f;

  const int lane = threadIdx.x & 31;
  const int wave = threadIdx.x >> 5;
  const int mn   = lane & 15;            // M (row l) for A, N (col) for B/D
  const int kbl  = (lane >> 4) << 1;     // K pair base: 0 or 2

  __syncthreads();

  for (int ch = 0; ch < (Ln + 15) / 16; ++ch) {
    const int l0 = ch * 16;

    // ---- stage 1: conv + silu (rank-1 input projection folded in) ----
    for (int l = 0; l < 16; ++l) {
      const int gl = l0 + l;
      float u = 0.f;
      if (gl < Ln) {
        float sv = 0.f, bsum = 0.f;
        for (int k = 0; k < DC; ++k) {
          const int idx = gl + k;          // xs[idx] == x_in[gl-3+k] scalar part
          sv += cw[k] * xs[idx];
          bsum += (idx >= 3) ? cw[k] : 0.f; // zero-padding excludes b_eff term
        }
        const float v = cb + beff_u * bsum + weff_u * sv;
        u = v / (1.f + __expf(-v));         // silu
      }
      u_buf[l][d] = u;
    }
    __syncthreads();

    // ---- stage 2: x_dbl[l][q] = sum_d u[l][d] * xp[q][d] via WMMA f32 16x16x4 ----
    if (wave < 3) {                        // wave-uniform branch: EXEC all ones
      v8f acc8 = {};
      const int q = wave * 16 + mn;
      const float* brow = xp[(q < 40) ? q : 0];  // cols >=40 compute garbage, never read
      for (int kk = 0; kk < DIN / 4; ++kk) {
        const int k0 = kk * 4 + kbl;
        v2f a, bf;
        a[0]  = u_buf[mn][k0]; a[1]  = u_buf[mn][k0 + 1];
        bf[0] = brow[k0];      bf[1] = brow[k0 + 1];
        acc8 = wmma_f32_16x16x4(a, bf, acc8);
      }
      const int mb = (lane >> 4) * 8;
      for (int r = 0; r < 8; ++r) xdbl[mb + r][q] = acc8[r];
    }
    __syncthreads();

    // ---- stage 3: dt + softplus, selective scan, silu(z) gate, mean accum ----
    for (int l = 0; l < 16; ++l) {
      const int gl = l0 + l;
      if (gl >= Ln) break;                 // uniform across block
      float dtr = dtb;
      for (int r = 0; r < DTR; ++r) dtr += xdbl[l][r] * dtw[r];
      const float dt = (dtr > 20.f) ? dtr : log1pf(__expf(dtr));
      const float u  = u_buf[l][d];
      const float du = dt * u;
      float y = 0.f;
      for (int s = 0; s < DS; ++s) {
        const float dA = __expf(dt * Amat[s]);
        hst[s] = hst[s] * dA + du * xdbl[l][DTR + s];      // B_t
        y += hst[s] * xdbl[l][DTR + DS + s];               // C_t
      }
      const float z   = xs[3 + gl] * weff_z + beff_z;      // rank-1 z-path
      const float sil = z / (1.f + __expf(-z));
      acc += (y + Dd * u) * sil;
    }
    __syncthreads();
  }

  y_mean[(b * Cn + c) * DIN + d] = acc * (1.f / (float)Ln);
}

// pooled[b][c*H+h] = sum_d y_mean[b][c][d] * out_proj_w[c][h][d]  (mean commuted past proj)
__global__ __launch_bounds__(256)
void k_pool(const float* __restrict__ y_mean, const float* __restrict__ out_proj_w,
            float* __restrict__ pooled)
{
  __shared__ float ym_s[16][68];
  __shared__ float op_s[128][68];
  const int c    = blockIdx.x;
  const int lane = threadIdx.x & 31;
  const int wave = threadIdx.x >> 5;
  const int mn   = lane & 15;
  const int kbl  = (lane >> 4) << 1;
  v8f acc8 = {};
  for (int kc = 0; kc < DIN / 64; ++kc) {
    const int kb0 = kc * 64;
    __syncthreads();
    for (int i = threadIdx.x; i < 128 * 64; i += 256) {
      const int hh = i >> 6, kk = i & 63;
      op_s[hh][kk] = out_proj_w[(c * Hn + hh) * DIN + kb0 + kk];
    }
    for (int i = threadIdx.x; i < 16 * 64; i += 256) {
      const int m = i >> 6, kk = i & 63;
      ym_s[m][kk] = y_mean[(m * Cn + c) * DIN + kb0 + kk];
    }
    __syncthreads();
    for (int t = 0; t < 16; ++t) {
      const int k0 = t * 4 + kbl;
      v2f a, bf;
      a[0]  = ym_s[mn][k0];              a[1]  = ym_s[mn][k0 + 1];
      bf[0] = op_s[wave * 16 + mn][k0];  bf[1] = op_s[wave * 16 + mn][k0 + 1];
      acc8 = wmma_f32_16x16x4(a, bf, acc8);
    }
  }
  const int mb = (lane >> 4) * 8;
  const int hc = wave * 16 + mn;
  for (int r = 0; r < 8; ++r)
    pooled[(mb + r) * MIX + c * Hn + hc] = acc8[r];
}

// Ypart[kseg][b][j] = sum_{k in segment} X[b][k]*W[j][k].
// K split over blockIdx.y for HBM-saturating parallelism; weight tiles are
// double-buffered in LDS via async global->LDS B128 copies (ASYNCcnt).
__global__ __launch_bounds__(256)
void k_mlp_part(const float* __restrict__ X, const float* __restrict__ W,
                float* __restrict__ Ypart, int K, int N, int npc)
{
  __shared__ alignas(16) float x_s[2][16][68];
  __shared__ alignas(16) float w_s[2][128][68];

  const int jbase = blockIdx.x * 128;
  const int kseg  = blockIdx.y;
  const int kc0   = kseg * npc;           // first 64-wide K chunk of this segment
  const int tid   = threadIdx.x;
  const int lane  = tid & 31;
  const int wave  = tid >> 5;
  const int mn    = lane & 15;
  const int kbl   = (lane >> 4) << 1;

  // copy geometry: each thread moves 16B; 256 threads = 4KB per issue
  const int crow = tid >> 4;              // 0..15
  const int ccol = (tid & 15) * 4;        // float col 0,4,...,60 (16B aligned)
  const unsigned long long xb = (unsigned long long)(uintptr_t)X;
  const unsigned long long wb = (unsigned long long)(uintptr_t)W;

  auto issue_chunk = [&](int kc, int bb) {
    const long kb0 = (long)kc * 64;
    async_copy_b128(lds_off(&x_s[bb][crow][ccol]), xb,
                    (unsigned int)(((long)crow * K + kb0 + ccol) * 4));
    for (int it = 0; it < 8; ++it) {      // 8 issues = 128x64 W tile
      const int jj = it * 16 + crow;
      async_copy_b128(lds_off(&w_s[bb][jj][ccol]), wb,
                      (unsigned int)((((long)(jbase + jj)) * K + kb0 + ccol) * 4));
    }
  };

  issue_chunk(kc0, 0);
  v8f acc8 = {};
  for (int t = 0; t < npc; ++t) {
    const int bb = t & 1;
    wait_async0();                         // own async loads for chunk t done
    __syncthreads();                       // everyone's chunk t resident in LDS
    if (t + 1 < npc) issue_chunk(kc0 + t + 1, bb ^ 1);  // overlap with compute
    for (int s = 0; s < 16; ++s) {
      const int k0 = s * 4 + kbl;
      v2f a, bf;
      a[0]  = x_s[bb][mn][k0];               a[1]  = x_s[bb][mn][k0 + 1];
      bf[0] = w_s[bb][wave * 16 + mn][k0];   bf[1] = w_s[bb][wave * 16 + mn][k0 + 1];
      acc8 = wmma_f32_16x16x4(a, bf, acc8);
    }
    __syncthreads();                       // all waves done reading bb before refill
  }

  const int mb = (lane >> 4) * 8;
  const int j  = jbase + wave * 16 + mn;
  for (int r = 0; r < 8; ++r)
    Ypart[((long)kseg * Bn + (mb + r)) * N + j] = acc8[r];
}

// Y[b][j] = relu?(sum_seg Ypart[seg][b][j] + bias[j])
__global__ __launch_bounds__(256)
void k_reduce(const float* __restrict__ part, const float* __restrict__ bias,
              float* __restrict__ Y, int N, int ksegs, int do_relu)
{
  const int idx = blockIdx.x * 256 + threadIdx.x;
  if (idx >= Bn * N) return;
  const int b = idx / N, j = idx % N;
  float acc = bias[j];
  for (int s = 0; s < ksegs; ++s) acc += part[((long)s * Bn + b) * N + j];
  if (do_relu) acc = fmaxf(acc, 0.f);
  Y[(long)b * N + j] = acc;
}

__global__ void k_head(const float* __restrict__ h2, const float* __restrict__ w3,
                       const float* __restrict__ b3, float* __restrict__ out)
{
  const int t = threadIdx.x;
  if (t >= Bn * NCn) return;
  const int b = t >> 2, n = t & 3;
  float acc = b3[n];
  for (int k = 0; k < MIX; ++k)
    acc += h2[b * MIX + k] * w3[n * MIX + k];
  out[b * NCn + n] = acc;
}

extern "C" void kernel_launch(void* const* d_in, const int* in_sizes, int n_in,
                              void* d_out, int out_size, void* d_ws, size_t ws_size,
                              hipStream_t stream) {
  (void)in_sizes; (void)n_in; (void)out_size; (void)ws_size;
  const float* x          = (const float*)d_in[0];
  const float* W_emb      = (const float*)d_in[1];
  const float* b_emb      = (const float*)d_in[2];
  const float* in_proj_w  = (const float*)d_in[3];
  const float* conv_w     = (const float*)d_in[4];
  const float* conv_b     = (const float*)d_in[5];
  const float* x_proj_w   = (const float*)d_in[6];
  const float* dt_proj_w  = (const float*)d_in[7];
  const float* dt_proj_b  = (const float*)d_in[8];
  const float* A_log      = (const float*)d_in[9];
  const float* Dv         = (const float*)d_in[10];
  const float* out_proj_w = (const float*)d_in[11];
  const float* w1 = (const float*)d_in[12];
  const float* b1 = (const float*)d_in[13];
  const float* w2 = (const float*)d_in[14];
  const float* b2 = (const float*)d_in[15];
  const float* w3 = (const float*)d_in[16];
  const float* b3 = (const float*)d_in[17];

  float* ws     = (float*)d_ws;
  float* y_mean = ws;                          // B*C*DIN          = 90112
  float* pooled = y_mean + Bn * Cn * DIN;      // B*MIX            = 45056
  float* h1v    = pooled + Bn * MIX;           // B*3MIX           = 135168
  float* h2v    = h1v + Bn * MIX3;             // B*MIX            = 45056
  float* part1  = h2v + Bn * MIX;              // KSEG1*B*3MIX     = 540672
  float* part2  = part1 + (long)KSEG1 * Bn * MIX3;  // KSEG2*B*MIX = 270336

  k_scan<<<Bn * Cn, 256, 0, stream>>>(x, W_emb, b_emb, in_proj_w, conv_w, conv_b,
                                      x_proj_w, dt_proj_w, dt_proj_b, A_log, Dv, y_mean);
  k_pool<<<Cn, 256, 0, stream>>>(y_mean, out_proj_w, pooled);

  k_mlp_part<<<dim3(MIX3 / 128, KSEG1), 256, 0, stream>>>(pooled, w1, part1, MIX, MIX3, (MIX / 64) / KSEG1);
  k_reduce<<<(Bn * MIX3 + 255) / 256, 256, 0, stream>>>(part1, b1, h1v, MIX3, KSEG1, 1);

  k_mlp_part<<<dim3(MIX / 128, KSEG2), 256, 0, stream>>>(h1v, w2, part2, MIX3, MIX, (MIX3 / 64) / KSEG2);
  k_reduce<<<(Bn * MIX + 255) / 256, 256, 0, stream>>>(part2, b2, h2v, MIX, KSEG2, 1);

  k_head<<<1, 64, 0, stream>>>(h2v, w3, b3, (float*)d_out);
}